// MDA_39247411151344
// MI455X (gfx1250) — compile-verified
//
#include <hip/hip_runtime.h>
#include <hip/hip_bf16.h>
#include <math.h>

typedef __attribute__((ext_vector_type(2))) float v2f;
typedef __attribute__((ext_vector_type(8))) float v8f;

// Problem dims (match reference)
constexpr int N_ENT = 20000;
constexpr int D     = 901;
constexpr int TWO_D = 1802;
constexpr int H1    = 1024;
constexpr int H2    = 512;
constexpr int H3    = 64;
constexpr int B_TR  = 100000;
constexpr int B_TE  = 25000;

// Workspace layout (float indices)
constexpr int WS_W34   = 0;      // 512 floats:  W3 @ W4
constexpr int WS_W234  = 512;    // 1024 floats: W2 @ (W3 @ W4)
constexpr int WS_WCOMB = 1536;   // 1802 floats: W1 @ (W2 W3 W4)  (wA | wB)
constexpr int WS_BCOMB = 3344;   // 1 float:     collapsed bias
constexpr int WS_SA    = 3360;   // 20000 floats: sA[e] = feature[e] . wA
constexpr int WS_SB    = 23360;  // 20000 floats: sB[e] = feature[e] . wB

// ---------------------------------------------------------------------------
// K1: w34 = W3 @ W4   (512x64 @ 64x1), plus bias term b3.W4 + b4
// ---------------------------------------------------------------------------
__global__ void k1_collapse34(const float* __restrict__ W3, const float* __restrict__ W4,
                              const float* __restrict__ b3, const float* __restrict__ b4,
                              float* __restrict__ ws) {
    int t = threadIdx.x;  // 0..511
    float acc = 0.f;
    #pragma unroll 8
    for (int j = 0; j < H3; ++j) acc = fmaf(W3[t * H3 + j], W4[j], acc);
    ws[WS_W34 + t] = acc;
    if (t == 0) {
        float tb = b4[0];
        for (int j = 0; j < H3; ++j) tb = fmaf(b3[j], W4[j], tb);
        ws[WS_BCOMB] = tb;
    }
}

// ---------------------------------------------------------------------------
// K2: w234 = W2 @ w34  (1024x512 @ 512x1); row 1024 folds b2.w34 into bias.
// One wave per row; lanes stride the dot then shuffle-reduce (wave32).
// ---------------------------------------------------------------------------
__global__ void k2_w234(const float* __restrict__ W2, const float* __restrict__ b2,
                        float* __restrict__ ws) {
    int lane = threadIdx.x & 31;
    int wave = threadIdx.x >> 5;
    int row  = blockIdx.x * (blockDim.x >> 5) + wave;
    if (row > H1) return;
    const float* w34 = ws + WS_W34;
    const float* src = (row < H1) ? (W2 + (size_t)row * H2) : b2;
    float acc = 0.f;
    for (int j = lane; j < H2; j += 32) acc = fmaf(src[j], w34[j], acc);
    #pragma unroll
    for (int off = 16; off > 0; off >>= 1) acc += __shfl_down(acc, off, 32);
    if (lane == 0) {
        if (row < H1) ws[WS_W234 + row] = acc;
        else          ws[WS_BCOMB] += acc;   // kernels are stream-ordered
    }
}

// ---------------------------------------------------------------------------
// K3: w_comb = W1 @ w234 (1802x1024 @ 1024x1); row 1802 folds b1.w234 into bias.
// ---------------------------------------------------------------------------
__global__ void k3_wcomb(const float* __restrict__ W1, const float* __restrict__ b1,
                         float* __restrict__ ws) {
    int lane = threadIdx.x & 31;
    int wave = threadIdx.x >> 5;
    int row  = blockIdx.x * (blockDim.x >> 5) + wave;
    if (row > TWO_D) return;
    const float* w234 = ws + WS_W234;
    const float* src  = (row < TWO_D) ? (W1 + (size_t)row * H1) : b1;
    float acc = 0.f;
    for (int j = lane; j < H1; j += 32) acc = fmaf(src[j], w234[j], acc);
    #pragma unroll
    for (int off = 16; off > 0; off >>= 1) acc += __shfl_down(acc, off, 32);
    if (lane == 0) {
        if (row < TWO_D) ws[WS_WCOMB + row] = acc;
        else             ws[WS_BCOMB] += acc;
    }
}

// ---------------------------------------------------------------------------
// K4 (dominant, HBM-bound): entity scores via V_WMMA_F32_16X16X4_F32.
//   GEMM: [20000 x 901] @ [901 x 2]  -> sA, sB   (N padded to 16 with zeros)
// One wave per 16-row tile; 1250 tiles exactly cover 20000 rows.
// A layout (ISA 7.12.2, 32-bit A 16x4): lane = M (mod 16); lanes 0-15 carry
// K0/K1 in v[0]/v[1], lanes 16-31 carry K2/K3. B mirrors with N = lane&15.
// C/D: VGPR r holds row r (lanes 0-15) and row r+8 (lanes 16-31), N = lane&15.
//
// Inner loop is fully branchless: the K=901 remainder is handled by a single
// masked tail step, and B-column masking is a post-load select (no exec-mask
// branches around loads).
// ---------------------------------------------------------------------------
__global__ void k4_scores(const float* __restrict__ feature, float* __restrict__ ws) {
    int lane = threadIdx.x & 31;
    int wave = threadIdx.x >> 5;
    int tile = blockIdx.x * (blockDim.x >> 5) + wave;
    if (tile >= N_ENT / 16) return;           // wave-uniform: EXEC stays all-1s

    const float* wA = ws + WS_WCOMB;          // column N=0
    const float* wB = ws + WS_WCOMB + D;      // column N=1

    const int g = lane >> 4;                  // lane half: selects K offset pair
    const int l = lane & 15;                  // M for A, N for B
    const int row = tile * 16 + l;
    const float* frow = feature + (size_t)row * D;
    __builtin_prefetch(frow, 0, 3);           // global_prefetch_b8 (near caches)

    // Valid address for every lane; unused columns zeroed by select below.
    const float* wsel = (l == 0) ? wA : wB;
    const bool bcol = (l < 2);

    v8f acc = {};

    // Main loop: kb in [0, 900), so k0 = kb + 2g, k1 = k0 + 1 are < 901
    // for both lane halves -> no bounds predicates anywhere.
    for (int kb = 0; kb + 4 <= D - 1; kb += 4) {     // kb = 0, 4, ..., 896
        const int k0 = kb + (g << 1);
        v2f a, b;
        a.x = frow[k0];
        a.y = frow[k0 + 1];
        const float b0 = wsel[k0];
        const float b1 = wsel[k0 + 1];
        b.x = bcol ? b0 : 0.f;                       // v_cndmask, no branch
        b.y = bcol ? b1 : 0.f;
        acc = __builtin_amdgcn_wmma_f32_16x16x4_f32(
            false, a, false, b, (short)0, acc, false, false);
    }

    // Tail: k = 900 only (valid for lane half g==0, x component). Load the
    // valid element unconditionally on all lanes, then mask by lane group.
    {
        const float alast = frow[D - 1];
        const float blast = wsel[D - 1];
        const bool v0 = (g == 0);
        v2f a, b;
        a.x = v0 ? alast : 0.f;  a.y = 0.f;
        b.x = (v0 && bcol) ? blast : 0.f;  b.y = 0.f;
        acc = __builtin_amdgcn_wmma_f32_16x16x4_f32(
            false, a, false, b, (short)0, acc, false, false);
    }

    // Write out columns N=0 (sA) and N=1 (sB): lanes 0,1,16,17 hold them.
    if (bcol) {
        float* dst = ws + ((l == 0) ? WS_SA : WS_SB);
        const int rbase = tile * 16 + (g << 3);
        #pragma unroll
        for (int v = 0; v < 8; ++v) dst[rbase + v] = acc[v];
    }
}

// ---------------------------------------------------------------------------
// K5: pair gather + sigmoid. Score tables (80 KB each) are L2-resident.
// d_out = [train scores (100000) | test scores (25000)]
// ---------------------------------------------------------------------------
__global__ void k5_pairs(const int* __restrict__ tr, const int* __restrict__ te,
                         const float* __restrict__ ws, float* __restrict__ out) {
    int p = blockIdx.x * blockDim.x + threadIdx.x;
    if (p >= B_TR + B_TE) return;
    const float* sA = ws + WS_SA;
    const float* sB = ws + WS_SB;
    const float bc = ws[WS_BCOMB];
    int i, j;
    if (p < B_TR) { i = tr[2 * p];           j = tr[2 * p + 1]; }
    else          { int q = p - B_TR; i = te[2 * q]; j = te[2 * q + 1]; }
    float x = sA[i] + sB[j] + bc;
    out[p] = 1.f / (1.f + expf(-x));
}

// ---------------------------------------------------------------------------
extern "C" void kernel_launch(void* const* d_in, const int* in_sizes, int n_in,
                              void* d_out, int out_size, void* d_ws, size_t ws_size,
                              hipStream_t stream) {
    const float* feature = (const float*)d_in[0];
    const float* W1 = (const float*)d_in[1];
    const float* b1 = (const float*)d_in[2];
    const float* W2 = (const float*)d_in[3];
    const float* b2 = (const float*)d_in[4];
    const float* W3 = (const float*)d_in[5];
    const float* b3 = (const float*)d_in[6];
    const float* W4 = (const float*)d_in[7];
    const float* b4 = (const float*)d_in[8];
    const int* tr = (const int*)d_in[9];
    const int* te = (const int*)d_in[10];
    float* ws  = (float*)d_ws;
    float* out = (float*)d_out;

    k1_collapse34<<<1, 512, 0, stream>>>(W3, W4, b3, b4, ws);

    // 1025 rows (incl. bias row), 8 waves/block of 256 threads
    k2_w234<<<(H1 + 1 + 7) / 8, 256, 0, stream>>>(W2, b2, ws);

    // 1803 rows (incl. bias row)
    k3_wcomb<<<(TWO_D + 1 + 7) / 8, 256, 0, stream>>>(W1, b1, ws);

    // 1250 tiles, one wave each
    k4_scores<<<(N_ENT / 16 + 7) / 8, 256, 0, stream>>>(feature, ws);

    k5_pairs<<<(B_TR + B_TE + 255) / 256, 256, 0, stream>>>(tr, te, ws, out);
}